// TripletLoss_84653805404173
// MI455X (gfx1250) — compile-verified
//
#include <hip/hip_runtime.h>
#include <math.h>

typedef __attribute__((ext_vector_type(2))) float v2f;
typedef __attribute__((ext_vector_type(8))) float v8f;

#define DIMS 512
#define NROWS 4096      // rows of the "n" half (reduction dim of the column reductions)
#define BT_STRIDE 516   // 512 + 4 dwords: 4*n distinct mod 64 -> conflict-free ds_load_b64

// ---------------- Kernel 1: row squared norms + accumulator init ----------------
__global__ __launch_bounds__(128) void norms_kernel(const float* __restrict__ feats,
                                                    float* __restrict__ norms,
                                                    float* __restrict__ acc) {
    int row = blockIdx.x;            // 0..8191
    int t   = threadIdx.x;           // 0..127, 4 elements each
    const float* src = feats + (size_t)row * DIMS + t * 4;
    float4 v = *(const float4*)src;
    float s = v.x * v.x + v.y * v.y + v.z * v.z + v.w * v.w;

    __shared__ float red[128];
    red[t] = s;
    __syncthreads();
    for (int off = 64; off > 0; off >>= 1) {
        if (t < off) red[t] += red[t + off];
        __syncthreads();
    }
    if (t == 0) {
        norms[row] = red[0];
        if (row == 0) { acc[0] = 0.0f; acc[1] = 0.0f; }   // runs before dist_kernel (stream order)
    }
}

// ---------------- Kernel 2: WMMA distance GEMM + column max/min reduction ----------------
__global__ __launch_bounds__(256) void dist_kernel(const float* __restrict__ feats,
                                                   const float* __restrict__ norms,
                                                   const int*   __restrict__ marginp,
                                                   float* __restrict__ acc) {
    const int jt    = blockIdx.x;        // 0..511 : 16-column tile of the 4096x8192 D matrix
    const int jbase = jt << 4;           // column base (also feats row index of Y)
    const bool is_n = (jt < 256);        // first 4096 cols: n_d (max); rest: a_d (min+hinge)

    __shared__ float Bt[16][BT_STRIDE];  // B tile transposed: Bt[n][k] = feats[jbase+n][k]
    __shared__ float Xn[NROWS];          // squared norms of all n rows
    __shared__ float Yn[16];             // squared norms of this column tile
    __shared__ float red[8][32];
    __shared__ float colres[16];

    const int t = threadIdx.x;

    // Stage B tile (coalesced: 16 consecutive threads read 16 consecutive k of one row)
    {
        int n  = t >> 4;
        int k0 = t & 15;
        const float* src = feats + (size_t)(jbase + n) * DIMS;
        #pragma unroll
        for (int k = k0; k < DIMS; k += 16) Bt[n][k] = src[k];
        if (t < 16) Yn[t] = norms[jbase + t];
    }
    // Stage x-norms
    for (int i = t; i < NROWS; i += 256) Xn[i] = norms[i];
    __syncthreads();

    const int wave = t >> 5;
    const int lane = t & 31;
    const int half = lane >> 4;          // h: lane group 0-15 vs 16-31
    const int col  = lane & 15;          // column within tile (B/C) == row m (A)

    float run = is_n ? 0.0f : 3.402823e38f;

    const float* bp = &Bt[col][2 * half];
    // Each wave sweeps quad i-tiles (64 rows) qt = wave, wave+8, ... : 64 quads = 4096 rows
    for (int qt = wave; qt < (NROWS / 64); qt += 8) {
        const int i0 = qt << 6;
        // Four A-fragment row streams: rows i0 + 16*q + m, K offset selected by half
        const float* a0 = feats + (size_t)(i0 + 0  + col) * DIMS + 2 * half;
        const float* a1 = feats + (size_t)(i0 + 16 + col) * DIMS + 2 * half;
        const float* a2 = feats + (size_t)(i0 + 32 + col) * DIMS + 2 * half;
        const float* a3 = feats + (size_t)(i0 + 48 + col) * DIMS + 2 * half;

        v8f c0 = {}, c1 = {}, c2 = {}, c3 = {};   // 4 independent f32 16x16 accumulators
        #pragma unroll 4
        for (int k = 0; k < DIMS; k += 4) {
            v2f b  = *(const v2f*)(bp + k);       // B[k+2h][n], B[k+2h+1][n] (shared)
            v2f f0 = *(const v2f*)(a0 + k);
            v2f f1 = *(const v2f*)(a1 + k);
            v2f f2 = *(const v2f*)(a2 + k);
            v2f f3 = *(const v2f*)(a3 + k);
            c0 = __builtin_amdgcn_wmma_f32_16x16x4_f32(false, f0, false, b, (short)0, c0, false, false);
            c1 = __builtin_amdgcn_wmma_f32_16x16x4_f32(false, f1, false, b, (short)0, c1, false, false);
            c2 = __builtin_amdgcn_wmma_f32_16x16x4_f32(false, f2, false, b, (short)0, c2, false, false);
            c3 = __builtin_amdgcn_wmma_f32_16x16x4_f32(false, f3, false, b, (short)0, c3, false, false);
        }

        // c*[v] = dot(n_{i0+16q+v+8h}, y_{jbase+n});  dist^2 = xn + yn - 2*dot
        const float yn = Yn[col];
        #pragma unroll
        for (int v = 0; v < 8; ++v) {
            int r = i0 + v + 8 * half;
            float s0 = fmaxf(Xn[r +  0] + yn - 2.0f * c0[v], 0.0f);
            float s1 = fmaxf(Xn[r + 16] + yn - 2.0f * c1[v], 0.0f);
            float s2 = fmaxf(Xn[r + 32] + yn - 2.0f * c2[v], 0.0f);
            float s3 = fmaxf(Xn[r + 48] + yn - 2.0f * c3[v], 0.0f);
            float d0 = sqrtf(s0), d1 = sqrtf(s1), d2 = sqrtf(s2), d3 = sqrtf(s3);
            if (is_n) {
                run = fmaxf(run, fmaxf(fmaxf(d0, d1), fmaxf(d2, d3)));
            } else {
                run = fminf(run, fminf(fminf(d0, d1), fminf(d2, d3)));
            }
        }
    }

    red[wave][lane] = run;
    __syncthreads();

    // Combine 8 waves x 2 lane-halves per column
    if (t < 16) {
        float r = is_n ? 0.0f : 3.402823e38f;
        #pragma unroll
        for (int w = 0; w < 8; ++w) {
            float v0 = red[w][t];
            float v1 = red[w][t + 16];
            r = is_n ? fmaxf(r, fmaxf(v0, v1)) : fminf(r, fminf(v0, v1));
        }
        colres[t] = r;
    }
    __syncthreads();

    if (t == 0) {
        float margin = (float)(*marginp);
        float s = 0.0f;
        #pragma unroll
        for (int n = 0; n < 16; ++n) {
            float v = colres[n];
            s += is_n ? v : fmaxf(0.0f, margin - v);
        }
        atomicAdd(&acc[is_n ? 0 : 1], s);   // global_atomic_add_f32
    }
}

// ---------------- Kernel 3: finalize ----------------
__global__ void finalize_kernel(const float* __restrict__ acc, float* __restrict__ out) {
    out[0] = acc[0] * (1.0f / 4096.0f) + acc[1] * (1.0f / 4096.0f);
}

extern "C" void kernel_launch(void* const* d_in, const int* in_sizes, int n_in,
                              void* d_out, int out_size, void* d_ws, size_t ws_size,
                              hipStream_t stream) {
    (void)in_sizes; (void)n_in; (void)out_size; (void)ws_size;
    const float* feats   = (const float*)d_in[0];   // 8192*512 f32
    const int*   marginp = (const int*)d_in[1];     // scalar int
    float* wsf   = (float*)d_ws;
    float* norms = wsf;                             // 8192 floats
    float* acc   = wsf + 8192;                      // 2 floats
    float* out   = (float*)d_out;

    norms_kernel<<<8192, 128, 0, stream>>>(feats, norms, acc);
    dist_kernel<<<512, 256, 0, stream>>>(feats, norms, marginp, acc);
    finalize_kernel<<<1, 1, 0, stream>>>(acc, out);
}